// LinkPredictor_90975997264359
// MI455X (gfx1250) — compile-verified
//
#include <hip/hip_runtime.h>

typedef float v2f __attribute__((ext_vector_type(2)));
typedef float v8f __attribute__((ext_vector_type(8)));

#define NN   50000            // nodes
#define NRL  12               // relations
#define NT   500000           // triples
#define HIDN 16               // hidden
#define EMBN 128              // embedding
#define RP   25               // 2*NRL+1
#define KDIM 400              // RP*HIDN (GEMM K)
#define NAUG (2*NT + NN)      // augmented edge count = 1,050,000

// Decode augmented edge e -> (s, r, o). Edge list: [fwd | inverse | self-loop]
__device__ __forceinline__ void aug_edge(const int* __restrict__ t, int e,
                                         int& s, int& r, int& o) {
    if (e < NT)            { s = t[3*e];     r = t[3*e+1];       o = t[3*e+2]; }
    else if (e < 2*NT)     { int j = e - NT;
                             s = t[3*j+2];   r = t[3*j+1] + NRL; o = t[3*j];   }
    else                   { int i = e - 2*NT;
                             s = i;          r = 2*NRL;          o = i;        }
}

__global__ void k_zero(float4* __restrict__ p, int n4) {
    int i = blockIdx.x * blockDim.x + threadIdx.x;
    if (i < n4) p[i] = make_float4(0.f, 0.f, 0.f, 0.f);
}

// deg[r*NN + o] += 1 over augmented edges
__global__ void k_deg(const int* __restrict__ t, float* __restrict__ deg) {
    int e = blockIdx.x * blockDim.x + threadIdx.x;
    if (e >= NAUG) return;
    int s, r, o; aug_edge(t, e, s, r, o); (void)s;
    atomicAdd(&deg[r * NN + o], 1.0f);
}

// norm = 1 / max(deg, 1)  (in place)
__global__ void k_norm(float* __restrict__ deg) {
    int i = blockIdx.x * blockDim.x + threadIdx.x;
    if (i < RP * NN) deg[i] = 1.0f / fmaxf(deg[i], 1.0f);
}

// x1[o, :] += W1[(r,s), :] * en   (layer-1 message passing)
__global__ void k_msg(const int* __restrict__ t, const float* __restrict__ W1,
                      const float* __restrict__ norm, float* __restrict__ x1) {
    int e = blockIdx.x * blockDim.x + threadIdx.x;
    if (e >= NAUG) return;
    int s, r, o; aug_edge(t, e, s, r, o);
    float en = norm[r * NN + o];
    const float4* w = (const float4*)(W1 + ((long)r * NN + s) * HIDN);
    float* dst = x1 + (long)o * HIDN;
#pragma unroll
    for (int q = 0; q < 4; q++) {
        float4 v = w[q];
        atomicAdd(dst + 4*q + 0, v.x * en);
        atomicAdd(dst + 4*q + 1, v.y * en);
        atomicAdd(dst + 4*q + 2, v.z * en);
        atomicAdd(dst + 4*q + 3, v.w * en);
    }
}

// h = relu(x1 + b1)  (in place)
__global__ void k_relu(float* __restrict__ x1, const float* __restrict__ b1) {
    int i = blockIdx.x * blockDim.x + threadIdx.x;
    if (i < NN * HIDN) x1[i] = fmaxf(x1[i] + b1[i & (HIDN - 1)], 0.0f);
}

// agg[o, r*16 + :] += h[s, :] * en   -- (node, r*h) layout => dense GEMM A-matrix
__global__ void k_agg(const int* __restrict__ t, const float* __restrict__ h,
                      const float* __restrict__ norm, float* __restrict__ agg) {
    int e = blockIdx.x * blockDim.x + threadIdx.x;
    if (e >= NAUG) return;
    int s, r, o; aug_edge(t, e, s, r, o);
    float en = norm[r * NN + o];
    const float4* hs = (const float4*)(h + (long)s * HIDN);
    float* dst = agg + (long)o * KDIM + r * HIDN;
#pragma unroll
    for (int q = 0; q < 4; q++) {
        float4 v = hs[q];
        atomicAdd(dst + 4*q + 0, v.x * en);
        atomicAdd(dst + 4*q + 1, v.y * en);
        atomicAdd(dst + 4*q + 2, v.z * en);
        atomicAdd(dst + 4*q + 3, v.w * en);
    }
}

// x2 (50000x128) = agg (50000x400) @ W2 (400x128) + b2, via V_WMMA_F32_16X16X4_F32.
// One wave per 16-row strip; 8 v8f accumulators cover all 128 output columns.
__global__ void k_gemm(const float* __restrict__ agg, const float* __restrict__ W2,
                       const float* __restrict__ b2, float* __restrict__ x2) {
    int wave = (blockIdx.x * blockDim.x + threadIdx.x) >> 5;
    int lane = threadIdx.x & 31;
    if (wave >= NN / 16) return;           // wave-uniform: EXEC stays all-ones
    int m0   = wave * 16;
    int half = lane >> 4;                  // 0: lanes 0-15, 1: lanes 16-31
    int lr   = lane & 15;

    const float* arow = agg + (long)(m0 + lr) * KDIM;

    v8f c[8];
#pragma unroll
    for (int nt = 0; nt < 8; nt++) {
        v8f z = {0.f, 0.f, 0.f, 0.f, 0.f, 0.f, 0.f, 0.f};
        c[nt] = z;
    }

    for (int k0 = 0; k0 < KDIM; k0 += 4) {
        // A 16x4 fp32: lanes 0-15 -> K = k0,k0+1 ; lanes 16-31 -> K = k0+2,k0+3
        int ka = k0 + 2 * half;
        v2f a;
        a.x = arow[ka];
        a.y = arow[ka + 1];
        const float* brow0 = W2 + (long)ka * EMBN + lr;        // row K=ka
        const float* brow1 = W2 + (long)(ka + 1) * EMBN + lr;  // row K=ka+1
#pragma unroll
        for (int nt = 0; nt < 8; nt++) {
            v2f b;
            b.x = brow0[nt * 16];
            b.y = brow1[nt * 16];
            c[nt] = __builtin_amdgcn_wmma_f32_16x16x4_f32(
                false, a, false, b, (short)0, c[nt], false, false);
        }
    }

    // C/D layout: VGPR v holds rows M = v (lanes 0-15) and M = v+8 (lanes 16-31)
#pragma unroll
    for (int nt = 0; nt < 8; nt++) {
        float bias = b2[nt * 16 + lr];
#pragma unroll
        for (int v = 0; v < 8; v++) {
            int row = m0 + v + 8 * half;
            x2[(long)row * EMBN + nt * 16 + lr] = c[nt][v] + bias;
        }
    }
}

// scores[i] = sum_e x2[s,e] * rel[p,e] * x2[o,e] ; one wave32 per triple
__global__ void k_score(const int* __restrict__ t, const float* __restrict__ x2,
                        const float* __restrict__ rel, float* __restrict__ out) {
    int gid  = blockIdx.x * blockDim.x + threadIdx.x;
    int w    = gid >> 5;
    int lane = gid & 31;
    if (w >= NT) return;
    int s = t[3*w], p = t[3*w+1], o = t[3*w+2];
    const float* xs = x2 + (long)s * EMBN;
    const float* xo = x2 + (long)o * EMBN;
    const float* rp = rel + (long)p * EMBN;
    float acc = 0.f;
#pragma unroll
    for (int q = 0; q < 4; q++) {
        int j = lane + 32 * q;
        acc += xs[j] * rp[j] * xo[j];
    }
#pragma unroll
    for (int off = 16; off > 0; off >>= 1) acc += __shfl_xor(acc, off, 32);
    if (lane == 0) out[w] = acc;
}

extern "C" void kernel_launch(void* const* d_in, const int* in_sizes, int n_in,
                              void* d_out, int out_size, void* d_ws, size_t ws_size,
                              hipStream_t stream) {
    const int*   triples = (const int*)  d_in[0];   // (NT,3)
    const float* W1      = (const float*)d_in[1];   // (RP,NN,16)
    const float* b1      = (const float*)d_in[2];   // (16)
    const float* W2      = (const float*)d_in[3];   // (RP,16,128) == (400,128) flat
    const float* b2      = (const float*)d_in[4];   // (128)
    const float* rel     = (const float*)d_in[5];   // (12,128)
    float*       out     = (float*)d_out;           // (NT)

    // Workspace layout (floats): deg/norm | x1(h) | agg | x2
    float* deg = (float*)d_ws;
    float* x1  = deg + (long)RP * NN;           // +1,250,000
    float* agg = x1  + (long)NN * HIDN;         // +  800,000
    float* x2  = agg + (long)NN * KDIM;         // +20,000,000
    long zf = (long)RP * NN + (long)NN * HIDN + (long)NN * KDIM; // accum regions
    int n4 = (int)(zf / 4);

    dim3 B(256);
    k_zero<<<(n4 + 255) / 256, B, 0, stream>>>((float4*)d_ws, n4);
    k_deg <<<(NAUG + 255) / 256, B, 0, stream>>>(triples, deg);
    k_norm<<<(RP * NN + 255) / 256, B, 0, stream>>>(deg);
    k_msg <<<(NAUG + 255) / 256, B, 0, stream>>>(triples, W1, deg, x1);
    k_relu<<<(NN * HIDN + 255) / 256, B, 0, stream>>>(x1, b1);
    k_agg <<<(NAUG + 255) / 256, B, 0, stream>>>(triples, x1, deg, agg);

    int nwaves = NN / 16;                       // 3125 strips, 8 waves/block
    k_gemm<<<(nwaves * 32 + 255) / 256, B, 0, stream>>>(agg, W2, b2, x2);

    long sthreads = (long)NT * 32;
    k_score<<<(int)((sthreads + 255) / 256), B, 0, stream>>>(triples, x2, rel, out);
}